// MambaBlock_16965120820094
// MI455X (gfx1250) — compile-verified
//
#include <hip/hip_runtime.h>
#include <cstdint>

// ---- problem constants (match reference) ----
#define DIMC     256
#define D_STATE  16
#define D_CONV   4
#define D_INNER  512
#define DT_RANK  16
#define BSZ      4
#define LSEQ     4096
#define MTOT     (BSZ * LSEQ)      // 16384 rows (b,l)
#define NXZ      1024              // 2*D_INNER
#define NXD      48                // DT_RANK + 2*D_STATE

typedef float v2f __attribute__((ext_vector_type(2)));
typedef float v8f __attribute__((ext_vector_type(8)));

// fp32 WMMA 16x16x4: A 16x4 (2 VGPR), B 4x16 (2 VGPR), C/D 16x16 f32 (8 VGPR)
// 8-arg form: (neg_a, A, neg_b, B, c_mod, C, reuse_a, reuse_b)
__device__ __forceinline__ v8f wmma_f32(v2f a, v2f b, v8f c) {
  return __builtin_amdgcn_wmma_f32_16x16x4_f32(false, a, false, b, (short)0, c,
                                               false, false);
}

__device__ __forceinline__ float silu(float v) {
  return v / (1.0f + __expf(-v));
}
__device__ __forceinline__ float softplus_add(float v, float b) {
  v += b;
  return (v > 20.0f) ? v : log1pf(__expf(v));
}

// ------------------------------------------------------------------
// Kernel 1: xz[m, 0:1024] = seq(m, 0:256) * in_proj_w^T
//   A[m][k] = x[b, k, l]  (m = b*4096 + l), strided loads along k.
//   One wave -> 16x64 output strip. 16384 waves.
// ------------------------------------------------------------------
__global__ __launch_bounds__(128) void k_in_proj(const float* __restrict__ x,
                                                 const float* __restrict__ W,
                                                 float* __restrict__ xz) {
  const int wid  = blockIdx.x * 4 + (threadIdx.x >> 5);
  const int lane = threadIdx.x & 31;
  const int mTile = wid >> 4;           // 0..1023
  const int nStrip = wid & 15;          // 0..15
  const int m0 = mTile * 16;
  const int n0 = nStrip * 64;
  const int b  = m0 >> 12;
  const int l0 = m0 & 4095;
  const int half = lane >> 4, lr = lane & 15;

  v8f c0 = {}, c1 = {}, c2 = {}, c3 = {};
  const float* xA = x + (size_t)b * DIMC * LSEQ + l0 + lr;  // + k*LSEQ

  for (int k0 = 0; k0 < DIMC; k0 += 4) {
    const int kk = k0 + half * 2;
    v2f a;
    a.x = xA[(size_t)kk * LSEQ];
    a.y = xA[(size_t)(kk + 1) * LSEQ];
    const float* bp = W + (size_t)(n0 + lr) * DIMC + kk;
    v2f b0 = *(const v2f*)(bp);
    v2f b1 = *(const v2f*)(bp + 16 * DIMC);
    v2f b2 = *(const v2f*)(bp + 32 * DIMC);
    v2f b3 = *(const v2f*)(bp + 48 * DIMC);
    c0 = wmma_f32(a, b0, c0);
    c1 = wmma_f32(a, b1, c1);
    c2 = wmma_f32(a, b2, c2);
    c3 = wmma_f32(a, b3, c3);
  }
  float* o = xz + (size_t)(m0 + half * 8) * NXZ + n0 + lr;
#pragma unroll
  for (int r = 0; r < 8; ++r) {
    o[(size_t)r * NXZ +  0] = c0[r];
    o[(size_t)r * NXZ + 16] = c1[r];
    o[(size_t)r * NXZ + 32] = c2[r];
    o[(size_t)r * NXZ + 48] = c3[r];
  }
}

// ------------------------------------------------------------------
// Kernel 2: depthwise causal conv(4) + bias + SiLU over u = xz[:, 0:512]
// ------------------------------------------------------------------
__global__ __launch_bounds__(256) void k_conv_silu(const float* __restrict__ xz,
                                                   const float* __restrict__ cw,
                                                   const float* __restrict__ cb,
                                                   float* __restrict__ uact) {
  const int idx = blockIdx.x * blockDim.x + threadIdx.x;
  if (idx >= MTOT * D_INNER) return;
  const int d = idx & (D_INNER - 1);
  const int m = idx >> 9;
  const int l = m & 4095;
  const float* urow = xz + (size_t)m * NXZ + d;
  float acc = cb[d];
#pragma unroll
  for (int j = 0; j < D_CONV; ++j) {
    const int lj = l - 3 + j;
    if (lj >= 0) acc = fmaf(cw[d * D_CONV + j], urow[(ptrdiff_t)(j - 3) * NXZ], acc);
  }
  uact[(size_t)m * D_INNER + d] = silu(acc);
}

// ------------------------------------------------------------------
// Kernel 3: x_dbl[m, 0:48] = u_act(m, 0:512) * x_proj_w^T    (N=48: 3 tiles)
// ------------------------------------------------------------------
__global__ __launch_bounds__(128) void k_x_proj(const float* __restrict__ uact,
                                                const float* __restrict__ W,
                                                float* __restrict__ xdbl) {
  const int wid  = blockIdx.x * 4 + (threadIdx.x >> 5);  // 0..1023 (m tile)
  const int lane = threadIdx.x & 31;
  const int half = lane >> 4, lr = lane & 15;
  const int m0 = wid * 16;

  v8f c0 = {}, c1 = {}, c2 = {};
  for (int k0 = 0; k0 < D_INNER; k0 += 4) {
    const int kk = k0 + half * 2;
    v2f a  = *(const v2f*)(uact + (size_t)(m0 + lr) * D_INNER + kk);
    const float* bp = W + (size_t)lr * D_INNER + kk;
    v2f b0 = *(const v2f*)(bp);
    v2f b1 = *(const v2f*)(bp + 16 * D_INNER);
    v2f b2 = *(const v2f*)(bp + 32 * D_INNER);
    c0 = wmma_f32(a, b0, c0);
    c1 = wmma_f32(a, b1, c1);
    c2 = wmma_f32(a, b2, c2);
  }
  float* o = xdbl + (size_t)(m0 + half * 8) * NXD + lr;
#pragma unroll
  for (int r = 0; r < 8; ++r) {
    o[(size_t)r * NXD +  0] = c0[r];
    o[(size_t)r * NXD + 16] = c1[r];
    o[(size_t)r * NXD + 32] = c2[r];
  }
}

// ------------------------------------------------------------------
// Kernel 4: dt[m, 0:512] = softplus(dt_low(m, 0:16) * dt_proj_w^T + bias)
//   K=16 -> 4 WMMA steps. One wave -> 16x64 strip. 8192 waves.
// ------------------------------------------------------------------
__global__ __launch_bounds__(128) void k_dt(const float* __restrict__ xdbl,
                                            const float* __restrict__ W,
                                            const float* __restrict__ bias,
                                            float* __restrict__ dt) {
  const int wid  = blockIdx.x * 4 + (threadIdx.x >> 5);
  const int lane = threadIdx.x & 31;
  const int mTile = wid >> 3;          // 0..1023
  const int nStrip = wid & 7;          // 0..7
  const int m0 = mTile * 16;
  const int n0 = nStrip * 64;
  const int half = lane >> 4, lr = lane & 15;

  v8f c0 = {}, c1 = {}, c2 = {}, c3 = {};
  for (int k0 = 0; k0 < DT_RANK; k0 += 4) {
    const int kk = k0 + half * 2;
    v2f a  = *(const v2f*)(xdbl + (size_t)(m0 + lr) * NXD + kk);
    const float* bp = W + (size_t)(n0 + lr) * DT_RANK + kk;
    v2f b0 = *(const v2f*)(bp);
    v2f b1 = *(const v2f*)(bp + 16 * DT_RANK);
    v2f b2 = *(const v2f*)(bp + 32 * DT_RANK);
    v2f b3 = *(const v2f*)(bp + 48 * DT_RANK);
    c0 = wmma_f32(a, b0, c0);
    c1 = wmma_f32(a, b1, c1);
    c2 = wmma_f32(a, b2, c2);
    c3 = wmma_f32(a, b3, c3);
  }
  const float bi0 = bias[n0 + lr +  0];
  const float bi1 = bias[n0 + lr + 16];
  const float bi2 = bias[n0 + lr + 32];
  const float bi3 = bias[n0 + lr + 48];
  float* o = dt + (size_t)(m0 + half * 8) * D_INNER + n0 + lr;
#pragma unroll
  for (int r = 0; r < 8; ++r) {
    o[(size_t)r * D_INNER +  0] = softplus_add(c0[r], bi0);
    o[(size_t)r * D_INNER + 16] = softplus_add(c1[r], bi1);
    o[(size_t)r * D_INNER + 32] = softplus_add(c2[r], bi2);
    o[(size_t)r * D_INNER + 48] = softplus_add(c3[r], bi3);
  }
}

// ------------------------------------------------------------------
// Kernel 5: selective scan. Half-wave (16 lanes) per (b,d) row; lane = n.
//   h = exp(dt*A)*h + (dt*u)*B ; y = sum_n h*C (shfl-xor reduce)
//   Writes y (+ u*D, * silu(z)) in place over dt.
// ------------------------------------------------------------------
__global__ __launch_bounds__(128) void k_scan(const float* __restrict__ xz,
                                              const float* __restrict__ uact,
                                              const float* __restrict__ xdbl,
                                              float* __restrict__ dty,
                                              const float* __restrict__ A_log,
                                              const float* __restrict__ Dvec) {
  const int wid  = blockIdx.x * 4 + (threadIdx.x >> 5);  // 0..1023
  const int lane = threadIdx.x & 31;
  const int g = lane >> 4;      // which row in this wave
  const int n = lane & 15;      // state index
  const int row = wid * 2 + g;  // 0..2047
  const int b = row >> 9;
  const int d = row & (D_INNER - 1);

  const float Aval = -expf(A_log[d * D_STATE + n]);
  const float Dd   = Dvec[d];

  const float* dtp = dty  + (size_t)b * LSEQ * D_INNER + d;
  const float* up  = uact + (size_t)b * LSEQ * D_INNER + d;
  const float* zp  = xz   + (size_t)b * LSEQ * NXZ + D_INNER + d;
  const float* Bp  = xdbl + (size_t)b * LSEQ * NXD + DT_RANK + n;
  const float* Cp  = Bp + D_STATE;
  float* yp = dty + (size_t)b * LSEQ * D_INNER + d;

  float h = 0.0f;
  for (int t = 0; t < LSEQ; ++t) {
    const float dt_v = dtp[(size_t)t * D_INNER];
    const float u_v  = up[(size_t)t * D_INNER];
    const float Bv   = Bp[(size_t)t * NXD];
    const float Cv   = Cp[(size_t)t * NXD];
    const float dA = __expf(dt_v * Aval);
    h = fmaf(dA, h, (dt_v * u_v) * Bv);
    float p = h * Cv;
    p += __shfl_xor(p, 8, 32);
    p += __shfl_xor(p, 4, 32);
    p += __shfl_xor(p, 2, 32);
    p += __shfl_xor(p, 1, 32);
    if (n == 0) {
      const float z_v = zp[(size_t)t * NXZ];
      float y = fmaf(Dd, u_v, p);
      y *= silu(z_v);
      yp[(size_t)t * D_INNER] = y;   // overwrite dt (already consumed)
    }
  }
}

// ------------------------------------------------------------------
// Kernel 6: out[b, c, l] = y(m, 0:512) * out_proj_w^T, stored transposed.
// ------------------------------------------------------------------
__global__ __launch_bounds__(128) void k_out_proj(const float* __restrict__ y,
                                                  const float* __restrict__ W,
                                                  float* __restrict__ out) {
  const int wid  = blockIdx.x * 4 + (threadIdx.x >> 5);
  const int lane = threadIdx.x & 31;
  const int mTile = wid >> 2;          // 0..1023
  const int nStrip = wid & 3;          // 0..3
  const int m0 = mTile * 16;
  const int n0 = nStrip * 64;
  const int half = lane >> 4, lr = lane & 15;

  v8f c0 = {}, c1 = {}, c2 = {}, c3 = {};
  for (int k0 = 0; k0 < D_INNER; k0 += 4) {
    const int kk = k0 + half * 2;
    v2f a  = *(const v2f*)(y + (size_t)(m0 + lr) * D_INNER + kk);
    const float* bp = W + (size_t)(n0 + lr) * D_INNER + kk;
    v2f b0 = *(const v2f*)(bp);
    v2f b1 = *(const v2f*)(bp + 16 * D_INNER);
    v2f b2 = *(const v2f*)(bp + 32 * D_INNER);
    v2f b3 = *(const v2f*)(bp + 48 * D_INNER);
    c0 = wmma_f32(a, b0, c0);
    c1 = wmma_f32(a, b1, c1);
    c2 = wmma_f32(a, b2, c2);
    c3 = wmma_f32(a, b3, c3);
  }
  const int mrow = m0 + half * 8;
  const int b = mrow >> 12;
  const int l = mrow & 4095;
  float* o = out + (size_t)b * DIMC * LSEQ + l;   // + c*LSEQ + r
#pragma unroll
  for (int r = 0; r < 8; ++r) {
    o[(size_t)(n0 + lr +  0) * LSEQ + r] = c0[r];
    o[(size_t)(n0 + lr + 16) * LSEQ + r] = c1[r];
    o[(size_t)(n0 + lr + 32) * LSEQ + r] = c2[r];
    o[(size_t)(n0 + lr + 48) * LSEQ + r] = c3[r];
  }
}

extern "C" void kernel_launch(void* const* d_in, const int* in_sizes, int n_in,
                              void* d_out, int out_size, void* d_ws, size_t ws_size,
                              hipStream_t stream) {
  (void)in_sizes; (void)n_in; (void)out_size; (void)ws_size;
  const float* x        = (const float*)d_in[0];
  const float* in_w     = (const float*)d_in[1];
  const float* conv_w   = (const float*)d_in[2];
  const float* conv_b   = (const float*)d_in[3];
  const float* xproj_w  = (const float*)d_in[4];
  const float* dtproj_w = (const float*)d_in[5];
  const float* dtproj_b = (const float*)d_in[6];
  const float* A_log    = (const float*)d_in[7];
  const float* Dvec     = (const float*)d_in[8];
  const float* out_w    = (const float*)d_in[9];
  float* out = (float*)d_out;

  // workspace layout (floats): xz | u_act | x_dbl | dt(->y)  ~= 137 MB
  float* xz   = (float*)d_ws;
  float* uact = xz   + (size_t)MTOT * NXZ;
  float* xdbl = uact + (size_t)MTOT * D_INNER;
  float* dty  = xdbl + (size_t)MTOT * NXD;

  k_in_proj <<<4096, 128, 0, stream>>>(x, in_w, xz);
  k_conv_silu<<<(MTOT * D_INNER) / 256, 256, 0, stream>>>(xz, conv_w, conv_b, uact);
  k_x_proj  <<<256, 128, 0, stream>>>(uact, xproj_w, xdbl);
  k_dt      <<<2048, 128, 0, stream>>>(xdbl, dtproj_w, dtproj_b, dty);
  k_scan    <<<256, 128, 0, stream>>>(xz, uact, xdbl, dty, A_log, Dvec);
  k_out_proj<<<1024, 128, 0, stream>>>(dty, out_w, out);
}